// PointNetModule_1305670058590
// MI455X (gfx1250) — compile-verified
//
#include <hip/hip_runtime.h>
#include <math.h>

typedef __attribute__((ext_vector_type(2))) float v2f;
typedef __attribute__((ext_vector_type(8))) float v8f;
typedef unsigned long long u64;
typedef unsigned int u32;

#define KNEI 10
#define R2 0.0625f            // RADIUS^2
#define FPS_THREADS 1024
#define FPS_PTS 16            // 16384 / 1024

__device__ __forceinline__ u64 shfl_xor_u64(u64 v, int mask) {
    u32 lo = (u32)v, hi = (u32)(v >> 32);
    lo = __shfl_xor(lo, mask, 32);
    hi = __shfl_xor(hi, mask, 32);
    return ((u64)hi << 32) | lo;
}

// ---------------------------------------------------------------------------
// Kernel 0: zero the output (harness poisons d_out; rows >= s must be 0).
// ---------------------------------------------------------------------------
__global__ void zero_out_kernel(float4* __restrict__ out, int n4) {
    int i = blockIdx.x * blockDim.x + threadIdx.x;
    if (i < n4) out[i] = make_float4(0.f, 0.f, 0.f, 0.f);
}

// ---------------------------------------------------------------------------
// Kernel 1: Farthest Point Sampling. One workgroup, everything in LDS/VGPRs.
// LDS: px/py/pz [n] each (192KB) + 33 u64 reduction slots.
// Each thread owns 16 points (coords + min_d in VGPRs).
// Argmax key: (f32bits << 32) | (0xFFFFFFFF - idx)  -> max => first-index tie-break.
// ---------------------------------------------------------------------------
__global__ void fps_kernel(const float* __restrict__ pos, int* __restrict__ idx,
                           int n, int s) {
    extern __shared__ char smem[];
    float* px = (float*)smem;
    float* py = px + n;
    float* pz = py + n;
    u64*   red = (u64*)(pz + n);   // [33]

    const int tid = threadIdx.x;
    for (int i = tid; i < n; i += FPS_THREADS) {
        px[i] = pos[3 * i + 0];
        py[i] = pos[3 * i + 1];
        pz[i] = pos[3 * i + 2];
    }
    __syncthreads();

    float lx[FPS_PTS], ly[FPS_PTS], lz[FPS_PTS], md[FPS_PTS];
    const int base = tid * FPS_PTS;
#pragma unroll
    for (int t = 0; t < FPS_PTS; ++t) {
        lx[t] = px[base + t];
        ly[t] = py[base + t];
        lz[t] = pz[base + t];
        md[t] = 3.4028235e38f;
    }

    const int lane = tid & 31;
    const int wid  = tid >> 5;
    int last = 0;

    for (int k = 0; k < s; ++k) {
        if (tid == 0) idx[k] = last;
        const float cx = px[last], cy = py[last], cz = pz[last];

        u64 best = 0;
#pragma unroll
        for (int t = 0; t < FPS_PTS; ++t) {
            float dx = lx[t] - cx, dy = ly[t] - cy, dz = lz[t] - cz;
            float d  = fmaf(dz, dz, fmaf(dy, dy, dx * dx));
            float m  = fminf(md[t], d);
            md[t] = m;
            u64 key = ((u64)__float_as_uint(m) << 32) |
                      (u32)(0xFFFFFFFFu - (u32)(base + t));
            best = (key > best) ? key : best;
        }
        // wave32 max
#pragma unroll
        for (int off = 16; off; off >>= 1) {
            u64 o = shfl_xor_u64(best, off);
            best = (o > best) ? o : best;
        }
        if (lane == 0) red[wid] = best;
        __syncthreads();
        if (wid == 0) {
            u64 b = red[lane];
#pragma unroll
            for (int off = 16; off; off >>= 1) {
                u64 o = shfl_xor_u64(b, off);
                b = (o > b) ? o : b;
            }
            if (lane == 0) red[32] = b;
        }
        __syncthreads();
        last = (int)(0xFFFFFFFFu - (u32)(red[32] & 0xFFFFFFFFull));
        // red[32] is only rewritten after the next round's barrier -> safe.
    }
}

// ---------------------------------------------------------------------------
// Kernel 2: per-center top-10 neighbors + rel vectors.
// 8 centers per workgroup (one wave each); pos staged once in LDS (192KB).
// Per lane: sorted 10-entry u64 list (d2bits<<32 | point_idx), guarded
// unrolled insertion; then 10-round wave shuffle-min merge of 32 lists.
// Writes rel as float4 (x,y,z,valid) to ws — valid in .w is safe because the
// padded W1 row multiplying .w is zero.
// ---------------------------------------------------------------------------
__global__ void knn_kernel(const float* __restrict__ pos, const int* __restrict__ idx,
                           float4* __restrict__ rel, int n) {
    extern __shared__ char smem[];
    float* px = (float*)smem;
    float* py = px + n;
    float* pz = py + n;

    const int tid = threadIdx.x;
    for (int i = tid; i < n; i += 256) {
        px[i] = pos[3 * i + 0];
        py[i] = pos[3 * i + 1];
        pz[i] = pos[3 * i + 2];
    }
    __syncthreads();

    const int lane = tid & 31;
    const int wid  = tid >> 5;
    const int ci   = blockIdx.x * 8 + wid;      // center index (< s)
    const int c    = idx[ci];
    const float cx = px[c], cy = py[c], cz = pz[c];

    u64 kq[KNEI];
#pragma unroll
    for (int t = 0; t < KNEI; ++t) kq[t] = ~0ull;

    for (int p = lane; p < n; p += 32) {
        float dx = px[p] - cx, dy = py[p] - cy, dz = pz[p] - cz;
        float d  = fmaf(dz, dz, fmaf(dy, dy, dx * dx));
        u64 key = ((u64)__float_as_uint(d) << 32) | (u32)p;
        if (key < kq[KNEI - 1]) {
            u64 cur = key;
#pragma unroll
            for (int t = 0; t < KNEI; ++t) {
                u64 mn = (cur < kq[t]) ? cur : kq[t];
                u64 mx = (cur < kq[t]) ? kq[t] : cur;
                kq[t] = mn;
                cur   = mx;
            }
        }
    }

    // merge 32 sorted lists: 10 rounds of wave-min; round r lands in lane r.
    u64 mykey = ~0ull;
#pragma unroll
    for (int r = 0; r < KNEI; ++r) {
        u64 cand = kq[0];
        u64 g = cand;
#pragma unroll
        for (int off = 16; off; off >>= 1) {
            u64 o = shfl_xor_u64(g, off);
            g = (o < g) ? o : g;
        }
        if (lane == r) mykey = g;
        if (cand == g) {                 // keys unique -> exactly one winner
#pragma unroll
            for (int t = 0; t < KNEI - 1; ++t) kq[t] = kq[t + 1];
            kq[KNEI - 1] = ~0ull;
        }
    }

    if (lane < KNEI) {
        u32 p = (u32)mykey;
        float d2 = __uint_as_float((u32)(mykey >> 32));
        // NOTE: reference uses pos[:s] rows (ci), not the FPS center, for rel.
        float ix = px[ci], iy = py[ci], iz = pz[ci];
        float valid = (d2 <= R2) ? 1.0f : 0.0f;
        rel[ci * KNEI + lane] = make_float4(px[p] - ix, py[p] - iy, pz[p] - iz, valid);
    }
}

// ---------------------------------------------------------------------------
// Kernel 3: fused MLP + masked max, f32 WMMA (V_WMMA_F32_16X16X4_F32).
// Workgroup = 8 waves = 8 centers = 80 rows. GEMM1: 20 WMMAs (K=4).
// GEMM2: 40 tiles x 16 accumulate-WMMAs (K=64) via LDS-staged h1.
// h never touches HBM; epilogue does bias + (-inf) mask + 10-way max.
// ---------------------------------------------------------------------------
#define H1STR 65      // pad 64->65: A-operand re-read is conflict-free
#define HBSTR 132     // pad 128->132: lanes 0-15 / 16-31 hit disjoint banks

__device__ __forceinline__ v8f wmma4(v2f a, v2f b, v8f c) {
    return __builtin_amdgcn_wmma_f32_16x16x4_f32(false, a, false, b,
                                                 (short)0, c, false, false);
}

__global__ void mlp_kernel(const float* __restrict__ relbuf,
                           const float* __restrict__ W1, const float* __restrict__ b1,
                           const float* __restrict__ W2, const float* __restrict__ b2,
                           float* __restrict__ out) {
    extern __shared__ char smem[];
    float* sW1  = (float*)smem;            // [4][64]  (row 3 zero-padded)
    float* sb1  = sW1 + 4 * 64;            // [64]
    float* sW2  = sb1 + 64;                // [64][128]
    float* sb2  = sW2 + 64 * 128;          // [128]
    float* sRel = sb2 + 128;               // [80][4]  (.w = valid flag)
    float* sH1  = sRel + 80 * 4;           // [80][H1STR]
    float* sHB  = sH1 + 80 * H1STR;        // [80][HBSTR]

    const int tid  = threadIdx.x;
    const int lane = tid & 31;
    const int wid  = tid >> 5;
    const int c0   = blockIdx.x * 8;       // first center of this block
    const int hl   = lane & 15;            // N / M lane within half
    const int hi   = lane >> 4;            // half select (K group 0/2)

    for (int i = tid; i < 3 * 64; i += 256) sW1[i] = W1[i];
    for (int i = tid; i < 64; i += 256) { sW1[3 * 64 + i] = 0.0f; sb1[i] = b1[i]; }
    for (int i = tid; i < 64 * 128; i += 256) sW2[i] = W2[i];
    for (int i = tid; i < 128; i += 256) sb2[i] = b2[i];
    for (int i = tid; i < 80 * 4; i += 256) sRel[i] = relbuf[c0 * 40 + i];
    __syncthreads();

    // ---- GEMM1: h1 = relu(rel @ W1p + b1); 5 M-tiles x 4 N-tiles ----------
    for (int t = wid; t < 20; t += 8) {
        const int m0 = (t >> 2) * 16;
        const int n0 = (t & 3) * 16;
        v2f a, b;
        a.x = sRel[(m0 + hl) * 4 + 2 * hi + 0];     // K = 2*hi
        a.y = sRel[(m0 + hl) * 4 + 2 * hi + 1];     // K = 2*hi + 1
        b.x = sW1[(2 * hi + 0) * 64 + n0 + hl];
        b.y = sW1[(2 * hi + 1) * 64 + n0 + hl];
        v8f c = {};
        c = wmma4(a, b, c);
#pragma unroll
        for (int r = 0; r < 8; ++r) {
            const int m  = m0 + r + hi * 8;
            const int nn = n0 + hl;
            float v = c[r] + sb1[nn];
            sH1[m * H1STR + nn] = (v > 0.0f) ? v : 0.0f;
        }
    }
    __syncthreads();

    // ---- GEMM2: h = h1 @ W2 + b2, mask invalid rows with -inf -------------
    {
        const int n0 = wid * 16;            // wave -> N tile (8 tiles = 128)
        for (int mt = 0; mt < 5; ++mt) {
            const int m0 = mt * 16;
            v8f c = {};
#pragma unroll
            for (int k = 0; k < 64; k += 4) {
                v2f a, b;
                a.x = sH1[(m0 + hl) * H1STR + k + 2 * hi + 0];
                a.y = sH1[(m0 + hl) * H1STR + k + 2 * hi + 1];
                b.x = sW2[(k + 2 * hi + 0) * 128 + n0 + hl];
                b.y = sW2[(k + 2 * hi + 1) * 128 + n0 + hl];
                c = wmma4(a, b, c);
            }
#pragma unroll
            for (int r = 0; r < 8; ++r) {
                const int m  = m0 + r + hi * 8;
                const int nn = n0 + hl;
                float v = c[r] + sb2[nn];
                if (sRel[m * 4 + 3] == 0.0f) v = -__builtin_inff();
                sHB[m * HBSTR + nn] = v;
            }
        }
    }
    __syncthreads();

    // ---- reduce: max over 10 neighbor rows per center ---------------------
    for (int q = tid; q < 8 * 128; q += 256) {
        const int cl  = q >> 7;
        const int col = q & 127;
        float best = -__builtin_inff();
#pragma unroll
        for (int j = 0; j < KNEI; ++j)
            best = fmaxf(best, sHB[(cl * KNEI + j) * HBSTR + col]);
        out[(c0 + cl) * 128 + col] = (best == -__builtin_inff()) ? 0.0f : best;
    }
}

// ---------------------------------------------------------------------------
extern "C" void kernel_launch(void* const* d_in, const int* in_sizes, int n_in,
                              void* d_out, int out_size, void* d_ws, size_t ws_size,
                              hipStream_t stream) {
    const float* pos = (const float*)d_in[0];
    // d_in[1] = batch (unused, all zeros)
    const float* W1 = (const float*)d_in[2];
    const float* b1 = (const float*)d_in[3];
    const float* W2 = (const float*)d_in[4];
    const float* b2 = (const float*)d_in[5];
    float* out = (float*)d_out;

    const int n = in_sizes[0] / 3;          // 16384
    const int s = n / 2;                    // 8192 (FPS_RATIO = 0.5)

    int*    idx = (int*)d_ws;
    size_t  relOff = ((size_t)s * sizeof(int) + 255) & ~(size_t)255;
    float4* rel = (float4*)((char*)d_ws + relOff);

    // 0) zero output (rows >= s stay 0; harness poisons d_out)
    const int n4 = out_size / 4;
    zero_out_kernel<<<(n4 + 255) / 256, 256, 0, stream>>>((float4*)out, n4);

    // 1) FPS: one workgroup, pos + reduction in LDS
    size_t ldsFps = (size_t)3 * n * sizeof(float) + 33 * sizeof(unsigned long long);
    fps_kernel<<<1, FPS_THREADS, ldsFps, stream>>>(pos, idx, n, s);

    // 2) top-10 neighbors per center (8 centers / block, pos staged in LDS)
    size_t ldsKnn = (size_t)3 * n * sizeof(float);
    knn_kernel<<<s / 8, 256, ldsKnn, stream>>>(pos, idx, rel, n);

    // 3) fused WMMA MLP + masked neighbor max
    size_t ldsMlp = (size_t)(4 * 64 + 64 + 64 * 128 + 128 + 80 * 4 +
                             80 * H1STR + 80 * HBSTR) * sizeof(float);
    mlp_kernel<<<s / 8, 256, ldsMlp, stream>>>((const float*)rel, W1, b1, W2, b2, out);
}